// SparseConvBlock_52810917871748
// MI455X (gfx1250) — compile-verified
//
#include <hip/hip_runtime.h>

typedef __attribute__((ext_vector_type(2))) float v2f;
typedef __attribute__((ext_vector_type(8))) float v8f;

namespace {
constexpr int kNVox = 100000;
constexpr int kK    = 27;
constexpr int kM    = 60000;
constexpr int kCin  = 64;
constexpr int kCout = 64;
constexpr float kEps = 1e-5f;

constexpr int kTileM      = 16;
constexpr int kWaves      = 6;                    // waves per block
constexpr int kThreads    = kWaves * 32;          // 192
constexpr int kTilesPerK  = kM / kTileM;          // 3750 (exact)
constexpr int kBlocksPerK = kTilesPerK / kWaves;  // 625  (exact -> no tail, EXEC all-ones)
constexpr int kTStride    = 68;  // padded row stride (floats) for both LDS tiles:
                                 // bank = (4*lane + k + base) % 64 -> all 64 banks hit once
}

__global__ void spconv_zero_f4(float4* __restrict__ p, int n4) {
  int i = blockIdx.x * blockDim.x + threadIdx.x;
  if (i < n4) p[i] = make_float4(0.f, 0.f, 0.f, 0.f);
}

__global__ __launch_bounds__(kThreads)
void spconv_wmma_f32(const float* __restrict__ feats,
                     const float* __restrict__ W,
                     const int*   __restrict__ in_idx,
                     const int*   __restrict__ out_idx,
                     float*       __restrict__ out) {
  __shared__ float sWT[kCout * kTStride];               // W[k] TRANSPOSED: [cout][cin]
  __shared__ float sA[kWaves][kTileM * kTStride];       // per-wave gathered tile
  __shared__ int   sOut[kWaves][kTileM];                // per-wave scatter rows

  const int tid  = threadIdx.x;
  const int wave = tid >> 5;
  const int lane = tid & 31;

  const int k   = blockIdx.x / kBlocksPerK;
  const int blk = blockIdx.x % kBlocksPerK;

  // ---- stage W[k] (64x64) into LDS transposed; b128 loads, scatter stores ----
  {
    const float4* Wk4 = (const float4*)(W + (size_t)k * (kCin * kCout));
    for (int i = tid; i < (kCin * kCout) / 4; i += kThreads) {
      float4 w = Wk4[i];                 // element (r = cin, c..c+3 = cout)
      int r = i >> 4;
      int c = (i & 15) << 2;
      sWT[(c + 0) * kTStride + r] = w.x;
      sWT[(c + 1) * kTStride + r] = w.y;
      sWT[(c + 2) * kTStride + r] = w.z;
      sWT[(c + 3) * kTStride + r] = w.w;
    }
  }

  // ---- per-wave: gather 16 rows of feats into LDS (scalar row base) ----
  const int tile = blk * kWaves + wave;                 // always < kTilesPerK
  const int m0   = tile * kTileM;
  const int* inI  = in_idx  + (size_t)k * kM + m0;
  const int* outI = out_idx + (size_t)k * kM + m0;

  const int inV = inI[lane & 15];        // all lanes load (dup hi half), coalesced
  if (lane < kTileM) sOut[wave][lane] = outI[lane];

#pragma unroll
  for (int r = 0; r < kTileM; ++r) {
    const int row = __builtin_amdgcn_readlane(inV, r);  // SGPR row index
    v2f v = *(const v2f*)(feats + (size_t)row * kCin + lane * 2);
    *(v2f*)(&sA[wave][r * kTStride + lane * 2]) = v;    // conflict-free banks
  }

  __syncthreads();                                      // sWT visible block-wide

  // ---- build 16 A-fragments (ISA 7.12.2: 32-bit A 16x4 layout) ----
  const int jn    = lane & 15;          // matrix row (A) / column (B,D)
  const int ksel  = (lane >> 4) << 1;   // lanes 0-15 -> K 0/1, lanes 16-31 -> K 2/3
  const int rbase = (lane >> 4) << 3;   // D rows: 0..7 (lo lanes) / 8..15 (hi lanes)

  v2f afrag[16];
#pragma unroll
  for (int kb = 0; kb < 16; ++kb)
    afrag[kb] = *(const v2f*)(&sA[wave][jn * kTStride + kb * 4 + ksel]);

  float* basep[8];
#pragma unroll
  for (int r = 0; r < 8; ++r) {
    int rowIdx = sOut[wave][rbase + r];
    basep[r] = out + (size_t)rowIdx * kCout + jn;       // nb*16 folds into IOFFSET
  }

  // ---- D(16x64) = A(16x64) * W[k](64x64), 4 N-blocks x 16 K-steps ----
#pragma unroll
  for (int nb = 0; nb < 4; ++nb) {
    v8f acc = {};
    const float* bcol = &sWT[(nb * 16 + jn) * kTStride];
#pragma unroll
    for (int kb = 0; kb < 16; ++kb) {
      // B 4x16: VGPR0 = K row (4kb+ksel), VGPR1 = next K row; contiguous in sWT
      v2f b = *(const v2f*)(bcol + kb * 4 + ksel);
      acc = __builtin_amdgcn_wmma_f32_16x16x4_f32(
          /*neg_a=*/false, afrag[kb], /*neg_b=*/false, b,
          /*c_mod=*/(short)0, acc, /*reuse_a=*/false, /*reuse_b=*/false);
    }
    // scatter-add the 16x16 block (D VGPR r -> rows rbase+r, col = nb*16+jn)
#pragma unroll
    for (int r = 0; r < 8; ++r)
      atomicAdd(basep[r] + nb * 16, acc[r]);
  }
}

__global__ void spconv_bn_relu(float* __restrict__ out,
                               const float* __restrict__ gamma,
                               const float* __restrict__ beta,
                               const float* __restrict__ mean,
                               const float* __restrict__ var,
                               int n4) {
  int i = blockIdx.x * blockDim.x + threadIdx.x;
  if (i >= n4) return;
  const int c0 = (i << 2) & (kCout - 1);
  float4 v = ((const float4*)out)[i];
  float s0 = __frsqrt_rn(var[c0 + 0] + kEps) * gamma[c0 + 0];
  float s1 = __frsqrt_rn(var[c0 + 1] + kEps) * gamma[c0 + 1];
  float s2 = __frsqrt_rn(var[c0 + 2] + kEps) * gamma[c0 + 2];
  float s3 = __frsqrt_rn(var[c0 + 3] + kEps) * gamma[c0 + 3];
  v.x = fmaxf((v.x - mean[c0 + 0]) * s0 + beta[c0 + 0], 0.f);
  v.y = fmaxf((v.y - mean[c0 + 1]) * s1 + beta[c0 + 1], 0.f);
  v.z = fmaxf((v.z - mean[c0 + 2]) * s2 + beta[c0 + 2], 0.f);
  v.w = fmaxf((v.w - mean[c0 + 3]) * s3 + beta[c0 + 3], 0.f);
  ((float4*)out)[i] = v;
}

extern "C" void kernel_launch(void* const* d_in, const int* in_sizes, int n_in,
                              void* d_out, int out_size, void* d_ws, size_t ws_size,
                              hipStream_t stream) {
  const float* feats   = (const float*)d_in[0];
  const float* W       = (const float*)d_in[1];
  const float* gamma   = (const float*)d_in[2];
  const float* beta    = (const float*)d_in[3];
  const float* rmean   = (const float*)d_in[4];
  const float* rvar    = (const float*)d_in[5];
  const int*   in_idx  = (const int*)d_in[6];
  const int*   out_idx = (const int*)d_in[7];
  float* out = (float*)d_out;

  const int n  = kNVox * kCout;    // 6.4M elems
  const int n4 = n / 4;

  spconv_zero_f4<<<(n4 + 255) / 256, 256, 0, stream>>>((float4*)out, n4);
  spconv_wmma_f32<<<kK * kBlocksPerK, kThreads, 0, stream>>>(feats, W, in_idx, out_idx, out);
  spconv_bn_relu<<<(n4 + 255) / 256, 256, 0, stream>>>(out, gamma, beta, rmean, rvar, n4);
}